// Density_estimator_850403524972
// MI455X (gfx1250) — compile-verified
//
#include <hip/hip_runtime.h>
#include <hip/hip_bf16.h>

// Problem sizes (fixed by the reference)
#define B_SZ 1024   // batch
#define F_SZ 256    // in_features
#define O_SZ 256    // out_features
#define D_SZ 2      // distributions
#define N_SZ (O_SZ * D_SZ)   // 512 GEMM columns (n = o*2 + d)
#define K_SZ (2 * F_SZ)      // 512 GEMM reduction dim ([x | x^2])
#define P_SZ (K_SZ / 2)      // 256 K-pairs (pair-interleaved layouts)

typedef float v2f __attribute__((ext_vector_type(2)));
typedef float v8f __attribute__((ext_vector_type(8)));

// ---------------------------------------------------------------------------
// Kernel 1: parameter preprocessing.
// Block n in [0,512) (n = o*D + d); thread f in [0,256).
//   w = 1 / (2 * softplus(rho)^2)
// Pair-interleaved K-major layout so WMMA B fragments are single b64 loads:
//   Hp[p][n][j] with p = k>>1, j = k&1   (flat: p*1024 + n*2 + j)
//   k = f      -> -2*w*m   (pairs with x   in G)
//   k = 256+f  ->  w       (pairs with x^2 in G)
//   C0[n] = sum_f w*m^2    (block LDS reduction)
// ---------------------------------------------------------------------------
__global__ void prep_params_kernel(const float* __restrict__ means,
                                   const float* __restrict__ rhos,
                                   float* __restrict__ Hp,
                                   float* __restrict__ C0) {
    __shared__ float red[F_SZ];
    const int n = blockIdx.x;
    const int f = threadIdx.x;
    const int idx = n * F_SZ + f;

    const float m   = means[idx];
    const float rho = rhos[idx];
    const float sig = (rho > 20.0f) ? rho : log1pf(__expf(rho));
    const float w   = 1.0f / (2.0f * sig * sig);

    const int p = f >> 1, j = f & 1;
    Hp[p * (N_SZ * 2) + n * 2 + j]                  = -2.0f * w * m;  // k = f
    Hp[(128 + p) * (N_SZ * 2) + n * 2 + j]          = w;              // k = 256+f

    red[f] = w * m * m;
    __syncthreads();
    #pragma unroll
    for (int s = F_SZ / 2; s > 0; s >>= 1) {
        if (f < s) red[f] += red[f + s];
        __syncthreads();
    }
    if (f == 0) C0[n] = red[0];
}

// ---------------------------------------------------------------------------
// Kernel 2: x preprocessing into pair-interleaved Gp[p][b][j]:
//   k = 2p+j;  Gp[p*2048 + b*2 + j] = x[b][k&255] (squared when k >= 256)
// Each thread handles one (p, b) pair: contiguous float2 read from x,
// fully coalesced float2 (b64) write to Gp.
// ---------------------------------------------------------------------------
__global__ void prep_x_kernel(const float* __restrict__ x,
                              float* __restrict__ Gp) {
    const int t = blockIdx.x * blockDim.x + threadIdx.x;  // [0, P_SZ*B_SZ)
    const int p = t >> 10;          // pair index [0,256)
    const int b = t & (B_SZ - 1);
    const int f0 = (p << 1) & (F_SZ - 1);   // both ks of a pair share the half
    float2 v = *(const float2*)(x + b * F_SZ + f0);
    if (p >= 128) { v.x *= v.x; v.y *= v.y; }   // x^2 half (k >= 256)
    *(float2*)(Gp + p * (B_SZ * 2) + b * 2) = v;
}

// ---------------------------------------------------------------------------
// Kernel 3: fused WMMA GEMM + epilogue.
//   est[b][n] = G @ H^T  (M=1024, N=512, K=512, f32) via V_WMMA_F32_16X16X4_F32
// Block = 128 threads = 4 waves; each wave owns a 32x32 output tile
// (2x2 subtiles of 16x16) -> each b64 fragment load feeds two WMMAs.
// Fragment layout (ISA 7.12.2, 16x4 f32):
//   A: lanes 0-15 -> M=lane, VGPRs = K={k0,k0+1}; lanes 16-31 -> K={k0+2,k0+3}
//   B: mirrored;  C/D: VGPR v -> M = v + 8*(lane>=16), N = lane&15
// Epilogue fused per tile: adjacent N columns are the d=0/d=1 pair, exchanged
// with __shfl_xor(1); even lanes compute exp/sum/max/ratio and store out[b,o].
// ---------------------------------------------------------------------------
__global__ void gemm_wmma_fused_kernel(const float* __restrict__ Gp,  // [P][B][2]
                                       const float* __restrict__ Hp,  // [P][N][2]
                                       const float* __restrict__ C0,  // [N]
                                       float* __restrict__ out) {     // [B][O]
    const int wave = threadIdx.x >> 5;
    const int lane = threadIdx.x & 31;
    const int half = lane >> 4;     // 0: K pair {k0,k0+1}; 1: {k0+2,k0+3}
    const int li   = lane & 15;

    const int rowBase = blockIdx.x * 128 + wave * 32;  // M tile base (32 rows)
    const int col0    = blockIdx.y * 32;               // N tile base (32 cols)

    v8f acc00 = {}, acc01 = {}, acc10 = {}, acc11 = {};

    #pragma unroll 2
    for (int k0 = 0; k0 < K_SZ; k0 += 4) {
        const int pidx = (k0 >> 1) + half;  // this half-wave's K pair
        // one b64 load per fragment, coalesced across each 16-lane half
        const v2f a0 = *(const v2f*)(Gp + pidx * (B_SZ * 2) + (rowBase + li) * 2);
        const v2f a1 = *(const v2f*)(Gp + pidx * (B_SZ * 2) + (rowBase + 16 + li) * 2);
        const v2f b0 = *(const v2f*)(Hp + pidx * (N_SZ * 2) + (col0 + li) * 2);
        const v2f b1 = *(const v2f*)(Hp + pidx * (N_SZ * 2) + (col0 + 16 + li) * 2);
        acc00 = __builtin_amdgcn_wmma_f32_16x16x4_f32(false, a0, false, b0,
                                                      (short)0, acc00, false, false);
        acc01 = __builtin_amdgcn_wmma_f32_16x16x4_f32(false, a0, false, b1,
                                                      (short)0, acc01, false, false);
        acc10 = __builtin_amdgcn_wmma_f32_16x16x4_f32(false, a1, false, b0,
                                                      (short)0, acc10, false, false);
        acc11 = __builtin_amdgcn_wmma_f32_16x16x4_f32(false, a1, false, b1,
                                                      (short)0, acc11, false, false);
    }

    // ---- fused epilogue ----------------------------------------------------
    // D layout: VGPR v -> row = subtileRow + v + 8*half, col n = subtileCol + li
    const bool evenLane = ((li & 1) == 0);
    #pragma unroll
    for (int nSub = 0; nSub < 2; ++nSub) {
        const int n  = col0 + nSub * 16 + li;
        const float c = C0[n];
        #pragma unroll
        for (int mSub = 0; mSub < 2; ++mSub) {
            const v8f acc = (mSub == 0) ? (nSub == 0 ? acc00 : acc01)
                                        : (nSub == 0 ? acc10 : acc11);
            #pragma unroll
            for (int v = 0; v < 8; ++v) {
                const float pr = __expf(-(acc[v] + c));     // probs for this n
                const float qr = __shfl_xor(pr, 1, 32);      // partner distr
                if (evenLane) {
                    const float s  = pr + qr;
                    const float mx = fmaxf(pr, qr);
                    const int row  = rowBase + mSub * 16 + v + 8 * half;
                    const int o    = n >> 1;
                    out[row * O_SZ + o] = s / (s + 2.0f - 2.0f * mx);
                }
            }
        }
    }
}

// ---------------------------------------------------------------------------
// Launcher. Inputs (setup_inputs order): x[B,F], means[O,D,F], rhos[O,D,F].
// Workspace layout (bytes):
//   Hp @ 0     : P_SZ*N_SZ*2*4 = 1 MiB
//   Gp @ 1 MiB : P_SZ*B_SZ*2*4 = 2 MiB
//   C0 @ 3 MiB : N_SZ*4        = 2 KiB
// ---------------------------------------------------------------------------
extern "C" void kernel_launch(void* const* d_in, const int* in_sizes, int n_in,
                              void* d_out, int out_size, void* d_ws, size_t ws_size,
                              hipStream_t stream) {
    const float* x     = (const float*)d_in[0];
    const float* means = (const float*)d_in[1];
    const float* rhos  = (const float*)d_in[2];
    float* out = (float*)d_out;

    char* ws = (char*)d_ws;
    float* Hp = (float*)(ws);
    float* Gp = (float*)(ws + (1u << 20));
    float* C0 = (float*)(ws + (3u << 20));

    // 1) parameter prep + C0 reduction
    prep_params_kernel<<<N_SZ, F_SZ, 0, stream>>>(means, rhos, Hp, C0);

    // 2) x prep into pair-interleaved Gp
    prep_x_kernel<<<(P_SZ * B_SZ) / 256, 256, 0, stream>>>(x, Gp);

    // 3) fused WMMA GEMM + epilogue -> out[B,O]
    dim3 ggrid(B_SZ / 128, N_SZ / 32);   // (8, 16)
    gemm_wmma_fused_kernel<<<ggrid, 128, 0, stream>>>(Gp, Hp, C0, out);
}